// VNTransformerEncoder_39092792328237
// MI455X (gfx1250) — compile-verified
//
#include <hip/hip_runtime.h>
#include <hip/hip_bf16.h>

typedef float v2f __attribute__((ext_vector_type(2)));
typedef float v8f __attribute__((ext_vector_type(8)));
typedef unsigned int v4u __attribute__((ext_vector_type(4)));
typedef int v4i __attribute__((ext_vector_type(4)));
typedef int v8i __attribute__((ext_vector_type(8)));

#define NTOK   2048            // tokens per batch
#define ROWS   12288           // 3 coord planes * 2 batches * 2048 tokens
#define BNROWS 4096            // 2 * 2048
#define DIMM   256
#define DINNER 512
#define DFF    1024
#define ATT_SCALE 0.072168783648703216f  // (3*64)^-0.5

static __device__ __forceinline__ v8f wmma_f32(v2f a, v2f b, v8f c) {
    return __builtin_amdgcn_wmma_f32_16x16x4_f32(false, a, false, b, (short)0, c,
                                                 false, false);
}

// ---------------------------------------------------------------------------
// TDM descriptor-based tile DMA: load a tile_rows x 32-float tile of a row-major
// [dimY, dimX] f32 tensor (row stride = dimX elements) into LDS, with hardware
// row padding of 4 DWORDs after every 32 DWORDs (matches LDS stride BK+4).
// D# layout per CDNA5 ISA ch.8 (group0: ctrl/lds/global/type, group1: dims).
// ---------------------------------------------------------------------------
static __device__ __forceinline__ void tdm_load_tile32(const float* gsrc,
                                                       unsigned lds_byte_off,
                                                       int dimX, int dimY,
                                                       int tile_rows) {
    unsigned long long ga = (unsigned long long)(uintptr_t)gsrc;
    v4u g0;
    g0[0] = 1u;                                   // count=1, is_restore=0
    g0[1] = lds_byte_off;                         // lds_addr
    g0[2] = (unsigned)(ga & 0xFFFFFFFFu);         // global_addr[31:0]
    g0[3] = (unsigned)((ga >> 32) & 0x01FFFFFFu)  // global_addr[56:32]
            | (2u << 30);                         // type = 2 ("image")
    v8i g1;
    g1[0] = (int)((2u << 16)      // data_size = 4 bytes
                | (1u << 20)      // pad_enable
                | (4u << 22)      // pad_interval: 32 DWORDs
                | (3u << 25));    // pad_amount: 4 DWORDs
    g1[1] = (int)(((unsigned)dimX & 0xFFFFu) << 16);                 // tensor_dim0 lo
    g1[2] = (int)((((unsigned)dimX >> 16) & 0xFFFFu)
                | (((unsigned)dimY & 0xFFFFu) << 16));               // dim0 hi | dim1 lo
    g1[3] = (int)((((unsigned)dimY >> 16) & 0xFFFFu) | (32u << 16)); // dim1 hi | tile_dim0=32
    g1[4] = tile_rows & 0xFFFF;                                      // tile_dim1 (tile_dim2=0)
    g1[5] = dimX;                                                    // tensor_dim0_stride lo
    g1[6] = 0;                                                       // stride0 hi | stride1 lo
    g1[7] = 0;
    v4i z4 = {0, 0, 0, 0};
    v8i z8 = {0, 0, 0, 0, 0, 0, 0, 0};
    __builtin_amdgcn_tensor_load_to_lds(g0, g1, z4, z4, z8, 0);
}

// ---------------------------------------------------------------------------
// GEMM: C[M,N] = A[M,K] @ W[N,K]^T   (vn_linear with c-planes flattened into M)
// Block tile 128x128, BK=32, 256 threads = 8 waves, wave tile 32x64.
// TDM double-buffered LDS staging; fp32 WMMA 16x16x4 inner product.
// ---------------------------------------------------------------------------
#define BM 128
#define BN 128
#define BK 32
#define BKP 36   // BK + 4 pad (TDM pads 4 DWORDs per 32)

__global__ __launch_bounds__(256) void gemm_xwt(const float* __restrict__ A,
                                                const float* __restrict__ W,
                                                float* __restrict__ C,
                                                int M, int N, int K) {
    __shared__ float As[2][BM][BKP];
    __shared__ float Bs[2][BN][BKP];

    const int tid  = threadIdx.x;
    const int wave = tid >> 5;
    const int lane = tid & 31;
    const int hl   = lane >> 4;      // half-wave selector
    const int r    = lane & 15;

    const int bm = blockIdx.y * BM;
    const int bn = blockIdx.x * BN;
    const int wm = (wave >> 1) * 32; // 4 waves along M
    const int wn = (wave & 1) * 64;  // 2 waves along N, 64 cols each

    v8f acc[2][4];
    for (int i = 0; i < 2; ++i)
        for (int j = 0; j < 4; ++j)
            acc[i][j] = (v8f){0.f,0.f,0.f,0.f,0.f,0.f,0.f,0.f};

    const unsigned ldsA0 = (unsigned)(uintptr_t)&As[0][0][0];
    const unsigned ldsA1 = (unsigned)(uintptr_t)&As[1][0][0];
    const unsigned ldsB0 = (unsigned)(uintptr_t)&Bs[0][0][0];
    const unsigned ldsB1 = (unsigned)(uintptr_t)&Bs[1][0][0];

    const int nkb = K / BK;
    if (wave == 0) {
        tdm_load_tile32(A + (size_t)bm * K, ldsA0, K, M, BM);
        tdm_load_tile32(W + (size_t)bn * K, ldsB0, K, N, BN);
    }

    for (int kb = 0; kb < nkb; ++kb) {
        const int cur = kb & 1;
        if (wave == 0) __builtin_amdgcn_s_wait_tensorcnt(0);
        __syncthreads();                       // buffers[cur] ready for all waves
        if (wave == 0 && (kb + 1) < nkb) {     // prefetch next K block into other buffer
            const int k0n = (kb + 1) * BK;
            tdm_load_tile32(A + (size_t)bm * K + k0n, cur ? ldsA0 : ldsA1, K, M, BM);
            tdm_load_tile32(W + (size_t)bn * K + k0n, cur ? ldsB0 : ldsB1, K, N, BN);
        }

        for (int kk = 0; kk < BK; kk += 4) {
            const int kc = kk + 2 * hl;
            v2f a0, a1, b0, b1, b2, b3;
            a0.x = As[cur][wm + r][kc];      a0.y = As[cur][wm + r][kc + 1];
            a1.x = As[cur][wm + 16 + r][kc]; a1.y = As[cur][wm + 16 + r][kc + 1];
            b0.x = Bs[cur][wn + r][kc];      b0.y = Bs[cur][wn + r][kc + 1];
            b1.x = Bs[cur][wn + 16 + r][kc]; b1.y = Bs[cur][wn + 16 + r][kc + 1];
            b2.x = Bs[cur][wn + 32 + r][kc]; b2.y = Bs[cur][wn + 32 + r][kc + 1];
            b3.x = Bs[cur][wn + 48 + r][kc]; b3.y = Bs[cur][wn + 48 + r][kc + 1];
            acc[0][0] = wmma_f32(a0, b0, acc[0][0]);
            acc[0][1] = wmma_f32(a0, b1, acc[0][1]);
            acc[0][2] = wmma_f32(a0, b2, acc[0][2]);
            acc[0][3] = wmma_f32(a0, b3, acc[0][3]);
            acc[1][0] = wmma_f32(a1, b0, acc[1][0]);
            acc[1][1] = wmma_f32(a1, b1, acc[1][1]);
            acc[1][2] = wmma_f32(a1, b2, acc[1][2]);
            acc[1][3] = wmma_f32(a1, b3, acc[1][3]);
        }
        __syncthreads();                       // all waves done reading buffers[cur]
    }

    // D layout: vgpr v -> row v + 8*hl, col r within each 16x16 tile
    for (int tm = 0; tm < 2; ++tm)
        for (int tn = 0; tn < 4; ++tn)
            for (int v = 0; v < 8; ++v) {
                int row = bm + wm + tm * 16 + v + 8 * hl;
                int col = bn + wn + tn * 16 + r;
                C[(size_t)row * N + col] = acc[tm][tn][v];
            }
}

// ---------------------------------------------------------------------------
// Flash attention: one wave per (b, h, 16-row i-tile). Effective K dim = 192
// (3 coords x 64 head dim) matching einsum('bhidc,bhjdc->bhij').
// ---------------------------------------------------------------------------
__global__ __launch_bounds__(32) void attn_kernel(const float* __restrict__ qkv,
                                                  float* __restrict__ att) {
    __shared__ float Qs[16][196];
    __shared__ float Ks[16][196];
    __shared__ float Vs[16][196];
    __shared__ float Ps[16][16];
    __shared__ float mrow[16], lrow[16], srow[16];

    const int blk = blockIdx.x;
    const int i0  = (blk & 127) * 16;
    const int h   = (blk >> 7) & 7;
    const int b   = blk >> 10;
    const int lane = threadIdx.x;
    const int hl = lane >> 4, r = lane & 15;
    const size_t QP = (size_t)BNROWS * 1536;

    for (int idx = lane; idx < 16 * 192; idx += 32) {
        int row = idx / 192, t = idx % 192, c = t / 64, d = t % 64;
        Qs[row][t] = qkv[c * QP + (size_t)(b * NTOK + i0 + row) * 1536 + h * 64 + d]
                     * ATT_SCALE;
    }
    if (lane < 16) { mrow[lane] = -3.0e38f; lrow[lane] = 0.f; }

    v8f o[12];
    for (int ch = 0; ch < 12; ++ch)
        o[ch] = (v8f){0.f,0.f,0.f,0.f,0.f,0.f,0.f,0.f};

    for (int j0 = 0; j0 < NTOK; j0 += 16) {
        __syncthreads();
        for (int idx = lane; idx < 16 * 192; idx += 32) {
            int row = idx / 192, t = idx % 192, c = t / 64, d = t % 64;
            size_t base = c * QP + (size_t)(b * NTOK + j0 + row) * 1536 + h * 64 + d;
            Ks[row][t] = qkv[base + 512];
            Vs[row][t] = qkv[base + 1024];
        }
        __syncthreads();

        v8f s = (v8f){0.f,0.f,0.f,0.f,0.f,0.f,0.f,0.f};
        for (int kk = 0; kk < 192; kk += 4) {
            const int kc = kk + 2 * hl;
            v2f a, bf;
            a.x  = Qs[r][kc]; a.y  = Qs[r][kc + 1];
            bf.x = Ks[r][kc]; bf.y = Ks[r][kc + 1];
            s = wmma_f32(a, bf, s);
        }
        for (int v = 0; v < 8; ++v) Ps[v + 8 * hl][r] = s[v];
        __syncthreads();

        if (lane < 16) {
            float mold = mrow[lane];
            float tmax = -3.0e38f;
            for (int c = 0; c < 16; ++c) tmax = fmaxf(tmax, Ps[lane][c]);
            float mnew = fmaxf(mold, tmax);
            float corr = __expf(mold - mnew);
            float sum = 0.f;
            for (int c = 0; c < 16; ++c) {
                float p = __expf(Ps[lane][c] - mnew);
                Ps[lane][c] = p;
                sum += p;
            }
            mrow[lane] = mnew;
            lrow[lane] = lrow[lane] * corr + sum;
            srow[lane] = corr;
        }
        __syncthreads();

        float sc[8];
        for (int v = 0; v < 8; ++v) sc[v] = srow[v + 8 * hl];
        for (int ch = 0; ch < 12; ++ch)
            for (int v = 0; v < 8; ++v) o[ch][v] *= sc[v];

        for (int ch = 0; ch < 12; ++ch) {
            const int t0 = ch * 16;
            for (int kk = 0; kk < 16; kk += 4) {
                const int kc = kk + 2 * hl;
                v2f a, bf;
                a.x  = Ps[r][kc];       a.y  = Ps[r][kc + 1];
                bf.x = Vs[kc][t0 + r];  bf.y = Vs[kc + 1][t0 + r];
                o[ch] = wmma_f32(a, bf, o[ch]);
            }
        }
    }
    __syncthreads();

    float inv[8];
    for (int v = 0; v < 8; ++v) inv[v] = 1.0f / lrow[v + 8 * hl];
    const size_t AP = (size_t)BNROWS * DINNER;
    for (int ch = 0; ch < 12; ++ch) {
        int t = ch * 16 + r, c = t / 64, d = t % 64;
        for (int v = 0; v < 8; ++v) {
            size_t row = (size_t)(b * NTOK + i0 + v + 8 * hl);
            att[c * AP + row * DINNER + h * 64 + d] = o[ch][v] * inv[v];
        }
    }
}

// ---------------------------------------------------------------------------
// VN layernorm (+ residual in-place into x): per (b,n) row of 256 features.
// ---------------------------------------------------------------------------
__global__ __launch_bounds__(256) void vn_ln_residual(const float* __restrict__ a,
                                                      const float* __restrict__ g,
                                                      float* __restrict__ x) {
    const int bn = blockIdx.x, d = threadIdx.x;
    const size_t PL = (size_t)BNROWS * DIMM;
    const size_t off = (size_t)bn * DIMM + d;
    float a0 = a[off], a1 = a[PL + off], a2 = a[2 * PL + off];
    float nrm = sqrtf(a0 * a0 + a1 * a1 + a2 * a2);

    __shared__ float r1[256], r2[256];
    r1[d] = nrm; r2[d] = nrm * nrm;
    __syncthreads();
    for (int st = 128; st > 0; st >>= 1) {
        if (d < st) { r1[d] += r1[d + st]; r2[d] += r2[d + st]; }
        __syncthreads();
    }
    float mu  = r1[0] * (1.f / 256.f);
    float var = r2[0] * (1.f / 256.f) - mu * mu;
    float ln  = (nrm - mu) * rsqrtf(var + 1e-5f) * g[d];
    float f   = ln / fmaxf(nrm, 1e-6f);
    x[off]          += a0 * f;
    x[PL + off]     += a1 * f;
    x[2 * PL + off] += a2 * f;
}

// ---------------------------------------------------------------------------
// VN-ReLU: elementwise over (bn, o) with 3-vector dot across coord planes.
// ---------------------------------------------------------------------------
__global__ __launch_bounds__(256) void vn_relu_kernel(const float* __restrict__ q,
                                                      const float* __restrict__ k,
                                                      float* __restrict__ out) {
    const size_t PL = (size_t)BNROWS * DFF;
    size_t i = (size_t)blockIdx.x * 256 + threadIdx.x;
    if (i >= PL) return;
    float q0 = q[i], q1 = q[i + PL], q2 = q[i + 2 * PL];
    float k0 = k[i], k1 = k[i + PL], k2 = k[i + 2 * PL];
    float qk = q0 * k0 + q1 * k1 + q2 * k2;
    float kn = fmaxf(sqrtf(k0 * k0 + k1 * k1 + k2 * k2), 1e-6f);
    float dn = qk / kn;
    bool keep = (qk >= 0.f);
    out[i]          = keep ? q0 : q0 - dn * k0;
    out[i + PL]     = keep ? q1 : q1 - dn * k1;
    out[i + 2 * PL] = keep ? q2 : q2 - dn * k2;
}

// ---------------------------------------------------------------------------
// Layout conversion: [b,n,d,c] <-> coord planes [c][b*n][d]
// ---------------------------------------------------------------------------
__global__ __launch_bounds__(256) void transpose_in(const float* __restrict__ x,
                                                    float* __restrict__ xw) {
    size_t idx = (size_t)blockIdx.x * 256 + threadIdx.x;       // over 3*4096*256
    const size_t PL = (size_t)BNROWS * DIMM;
    if (idx >= 3 * PL) return;
    int c = (int)(idx / PL);
    size_t rem = idx % PL;
    size_t bn = rem / DIMM;
    int d = (int)(rem % DIMM);
    xw[idx] = x[bn * (DIMM * 3) + (size_t)d * 3 + c];
}

__global__ __launch_bounds__(256) void transpose_out(const float* __restrict__ xw,
                                                     float* __restrict__ out) {
    size_t idx = (size_t)blockIdx.x * 256 + threadIdx.x;       // over 4096*256*3
    const size_t PL = (size_t)BNROWS * DIMM;
    if (idx >= 3 * PL) return;
    size_t bn = idx / (DIMM * 3);
    int rr = (int)(idx % (DIMM * 3));
    int d = rr / 3, c = rr % 3;
    out[idx] = xw[(size_t)c * PL + bn * DIMM + d];
}

// ---------------------------------------------------------------------------
extern "C" void kernel_launch(void* const* d_in, const int* in_sizes, int n_in,
                              void* d_out, int out_size, void* d_ws, size_t ws_size,
                              hipStream_t stream) {
    (void)in_sizes; (void)n_in; (void)out_size; (void)ws_size;
    const float* x    = (const float*)d_in[0];
    const float* Wqkv = (const float*)d_in[1];
    const float* Wout = (const float*)d_in[2];
    const float* g1   = (const float*)d_in[3];
    const float* Wff1 = (const float*)d_in[4];
    const float* Wr   = (const float*)d_in[5];
    const float* Ur   = (const float*)d_in[6];
    const float* Wff2 = (const float*)d_in[7];
    const float* g2   = (const float*)d_in[8];
    float* out = (float*)d_out;

    float* ws  = (float*)d_ws;
    float* xw  = ws;                                  // 12288*256
    float* qkv = xw  + (size_t)ROWS * DIMM;           // 12288*1536
    float* att = qkv + (size_t)ROWS * 3 * DINNER;     // 12288*512
    float* ab  = att + (size_t)ROWS * DINNER;         // 12288*256
    float* hb  = ab  + (size_t)ROWS * DIMM;           // 12288*1024
    float* qb  = hb  + (size_t)ROWS * DFF;            // 12288*1024
    float* kb  = qb  + (size_t)ROWS * DFF;            // 12288*1024

    const int elem_blocks = (int)(((size_t)ROWS * DIMM + 255) / 256);  // 12288
    transpose_in<<<elem_blocks, 256, 0, stream>>>(x, xw);

    for (int i = 0; i < 4; ++i) {
        const float* wqkv = Wqkv + (size_t)i * 1536 * DIMM;
        const float* wout = Wout + (size_t)i * DIMM * DINNER;
        const float* wf1  = Wff1 + (size_t)i * DFF * DIMM;
        const float* wr   = Wr   + (size_t)i * DFF * DFF;
        const float* ur   = Ur   + (size_t)i * DFF * DFF;
        const float* wf2  = Wff2 + (size_t)i * DIMM * DFF;

        // attention block
        gemm_xwt<<<dim3(1536 / BN, ROWS / BM), 256, 0, stream>>>(
            xw, wqkv, qkv, ROWS, 1536, DIMM);
        attn_kernel<<<2048, 32, 0, stream>>>(qkv, att);
        gemm_xwt<<<dim3(DIMM / BN, ROWS / BM), 256, 0, stream>>>(
            att, wout, ab, ROWS, DIMM, DINNER);
        vn_ln_residual<<<BNROWS, 256, 0, stream>>>(ab, g1 + (size_t)i * DIMM, xw);

        // feed-forward block
        gemm_xwt<<<dim3(DFF / BN, ROWS / BM), 256, 0, stream>>>(
            xw, wf1, hb, ROWS, DFF, DIMM);
        gemm_xwt<<<dim3(DFF / BN, ROWS / BM), 256, 0, stream>>>(
            hb, wr, qb, ROWS, DFF, DFF);
        gemm_xwt<<<dim3(DFF / BN, ROWS / BM), 256, 0, stream>>>(
            hb, ur, kb, ROWS, DFF, DFF);
        vn_relu_kernel<<<(int)(((size_t)BNROWS * DFF + 255) / 256), 256, 0, stream>>>(
            qb, kb, hb);
        gemm_xwt<<<dim3(DIMM / BN, ROWS / BM), 256, 0, stream>>>(
            hb, wf2, ab, ROWS, DIMM, DFF);
        vn_ln_residual<<<BNROWS, 256, 0, stream>>>(ab, g2 + (size_t)i * DIMM, xw);
    }

    transpose_out<<<elem_blocks, 256, 0, stream>>>(xw, out);
}